// NTM_12841952215819
// MI455X (gfx1250) — compile-verified
//
#include <hip/hip_runtime.h>
#include <math.h>

#define B_   256
#define N_   2048
#define M_   64
#define C_   512
#define R_   4
#define WH_  4
#define PR_  70
#define PW_  198
#define DE_  256
#define DO_  256
#define EPS_ 1e-8f

typedef float v2f __attribute__((ext_vector_type(2)));
typedef float v8f __attribute__((ext_vector_type(8)));

__device__ __forceinline__ float sigmoidf_(float x) { return 1.0f / (1.0f + expf(-x)); }
__device__ __forceinline__ float softplusf_(float x) { return (x > 20.0f) ? x : log1pf(expf(x)); }

__device__ __forceinline__ float wred(float v) {
#pragma unroll
  for (int o = 16; o > 0; o >>= 1) v += __shfl_xor(v, o, 32);
  return v;
}

// ---------------------------------------------------------------------------
// Generic fp32 WMMA GEMM: C[M,N] = [A1 | A2][M,Ktot] @ W[Ktot,N] + bias
// One wave per 16x16 C tile, K stepped by 4 via v_wmma_f32_16x16x4_f32.
// A frag: lane<16 -> row=lane, K={k0,k0+1}; lane>=16 -> row=lane-16,
// K={k0+2,k0+3} (one float2 load per lane).  B frag mirrored on columns,
// masked branch-free for the N edge.  C/D frag: vgpr i, lane l ->
// row = i + 8*(l>>4), col = l&15.
// K1 and Ktot must be multiples of 4 (true for all shapes used here), so the
// A1/A2 concat split is handled by two uniform loops (no divergence).
// ---------------------------------------------------------------------------
__global__ void wmma_gemm(const float* __restrict__ A1, int lda1, int K1,
                          const float* __restrict__ A2, int lda2,
                          const float* __restrict__ W, int ldw,
                          const float* __restrict__ bias,
                          float* __restrict__ Co, int ldc,
                          int Mrows, int Ncols, int Ktot) {
  const int lane = threadIdx.x & 31;
  const int half = lane >> 4;
  const int l15  = lane & 15;
  const int row0 = blockIdx.y << 4;
  const int col0 = blockIdx.x << 4;
  const int r = row0 + l15;
  const int c = col0 + l15;
  const bool cok = (c < Ncols);
  const int cc = cok ? c : (Ncols - 1);      // clamped, always in-bounds
  const float bmask = cok ? 1.0f : 0.0f;     // zero out ghost columns

  const float* __restrict__ wPtr = W + cc + (size_t)(2 * half) * ldw;
  const float* __restrict__ aPtr = A1 + (size_t)r * lda1 + 2 * half;
  v8f acc = {};

#pragma unroll 4
  for (int k0 = 0; k0 < K1; k0 += 4) {
    v2f av = *(const v2f*)aPtr;              // global_load_b64
    v2f bv;
    bv.x = wPtr[0] * bmask;
    bv.y = wPtr[ldw] * bmask;
    acc = __builtin_amdgcn_wmma_f32_16x16x4_f32(false, av, false, bv, (short)0, acc,
                                                false, false);
    aPtr += 4;
    wPtr += (size_t)4 * ldw;
  }
  if (K1 < Ktot) {                           // uniform branch: concat tail in A2
    const float* __restrict__ a2p = A2 + (size_t)r * lda2 + 2 * half;
#pragma unroll 4
    for (int k0 = K1; k0 < Ktot; k0 += 4) {
      v2f av = *(const v2f*)a2p;
      v2f bv;
      bv.x = wPtr[0] * bmask;
      bv.y = wPtr[ldw] * bmask;
      acc = __builtin_amdgcn_wmma_f32_16x16x4_f32(false, av, false, bv, (short)0, acc,
                                                  false, false);
      a2p += 4;
      wPtr += (size_t)4 * ldw;
    }
  }

  if (cok) {
    float bb = bias ? bias[c] : 0.0f;
#pragma unroll
    for (int i = 0; i < 8; ++i) {
      int rr = row0 + i + half * 8;
      if (rr < Mrows) Co[(size_t)rr * ldc + c] = acc[i] + bb;
    }
  }
}

// ---------------------------------------------------------------------------
// Per-(head,batch) scalars: k_norm, beta, g, softmax(s), gamma; write heads
// additionally erase=sigmoid, add=tanh vectors.  slots 0..3 read, 4..7 write.
// ---------------------------------------------------------------------------
__global__ void head_params(const float* __restrict__ rp, const float* __restrict__ wp,
                            float* __restrict__ prm, float* __restrict__ erase,
                            float* __restrict__ addv) {
  int slot = blockIdx.x;   // 0..7
  int b = threadIdx.x;     // 0..255
  const float* p = (slot < 4) ? (rp + (size_t)(slot * B_ + b) * PR_)
                              : (wp + (size_t)((slot - 4) * B_ + b) * PW_);
  float kn2 = 0.f;
  for (int m = 0; m < M_; ++m) { float v = p[m]; kn2 += v * v; }
  float s0 = p[M_ + 2], s1 = p[M_ + 3], s2 = p[M_ + 4];
  float mx = fmaxf(s0, fmaxf(s1, s2));
  float e0 = expf(s0 - mx), e1 = expf(s1 - mx), e2 = expf(s2 - mx);
  float es = e0 + e1 + e2;
  float* q = prm + (size_t)(slot * B_ + b) * 8;
  q[0] = sqrtf(kn2);
  q[1] = softplusf_(p[M_]);
  q[2] = sigmoidf_(p[M_ + 1]);
  q[3] = e0 / es; q[4] = e1 / es; q[5] = e2 / es;
  q[6] = 1.0f + softplusf_(p[M_ + 5]);
  q[7] = 0.f;
  if (slot >= 4) {
    int h = slot - 4;
    float* eo = erase + (size_t)(h * B_ + b) * M_;
    float* ao = addv + (size_t)(h * B_ + b) * M_;
    for (int m = 0; m < M_; ++m) {
      eo[m] = sigmoidf_(p[PR_ + m]);
      ao[m] = tanhf(p[PR_ + M_ + m]);
    }
  }
}

// ---------------------------------------------------------------------------
// Pass over ORIGINAL memory: cosine for 4 read heads + write head 0.
// Block per batch b, 8 waves; wave owns rows n = wave + 8*t; lane holds
// m = {2*lane, 2*lane+1}.  One 134MB read total; prefetch 64 rows ahead.
// ---------------------------------------------------------------------------
__global__ void cos_pass(const float* __restrict__ mem, const float* __restrict__ rp,
                         const float* __restrict__ wp, const float* __restrict__ prm,
                         float* __restrict__ cosR, float* __restrict__ cosW0) {
  int b = blockIdx.x;
  int lane = threadIdx.x & 31, wave = threadIdx.x >> 5;
  float2 kv[5]; float kn[5];
#pragma unroll
  for (int h = 0; h < 4; ++h) {
    kv[h] = *(const float2*)(rp + (size_t)(h * B_ + b) * PR_ + 2 * lane);
    kn[h] = prm[(size_t)(h * B_ + b) * 8];
  }
  kv[4] = *(const float2*)(wp + (size_t)b * PW_ + 2 * lane);
  kn[4] = prm[(size_t)(4 * B_ + b) * 8];
  const float2* mb = (const float2*)(mem + (size_t)b * N_ * M_);
  for (int n = wave; n < N_; n += 8) {
    __builtin_prefetch(mb + (size_t)(n + 64) * 32 + lane, 0, 0);
    float2 mv = mb[(size_t)n * 32 + lane];
    float nr = mv.x * mv.x + mv.y * mv.y;
    float d[5];
#pragma unroll
    for (int h = 0; h < 5; ++h) d[h] = mv.x * kv[h].x + mv.y * kv[h].y;
    nr = wred(nr);
#pragma unroll
    for (int h = 0; h < 5; ++h) d[h] = wred(d[h]);
    if (lane == 0) {
      float mn = sqrtf(nr);
#pragma unroll
      for (int h = 0; h < 4; ++h)
        cosR[((size_t)h * B_ + b) * N_ + n] = d[h] / (mn * kn[h] + EPS_);
      cosW0[(size_t)b * N_ + n] = d[4] / (mn * kn[4] + EPS_);
    }
  }
}

// ---------------------------------------------------------------------------
// Addressing: softmax(beta*cos) -> gate -> circular shift -> sharpen.
// Block per (b, h=blockIdx.y); in-place cos -> w.
// ---------------------------------------------------------------------------
__global__ void address_kernel(const float* __restrict__ cosBase,
                               const float* __restrict__ wPrevBase,
                               const float* __restrict__ prmBase,
                               float* __restrict__ wOutBase) {
  __shared__ float sh[N_];
  __shared__ float red[256];
  int b = blockIdx.x, h = blockIdx.y, tid = threadIdx.x;
  const float* cosp = cosBase + ((size_t)h * B_ + b) * N_;
  const float* wprev = wPrevBase + ((size_t)h * B_ + b) * N_;
  const float* q = prmBase + (size_t)(h * B_ + b) * 8;
  float* wout = wOutBase + ((size_t)h * B_ + b) * N_;
  float beta = q[1], g = q[2], s0 = q[3], s1 = q[4], s2 = q[5], gamma = q[6];

  float x[8];
  float mx = -INFINITY;
#pragma unroll
  for (int i = 0; i < 8; ++i) {
    int idx = tid + 256 * i;
    float v = beta * cosp[idx];
    x[i] = v; mx = fmaxf(mx, v);
  }
  red[tid] = mx; __syncthreads();
  for (int s = 128; s > 0; s >>= 1) { if (tid < s) red[tid] = fmaxf(red[tid], red[tid + s]); __syncthreads(); }
  mx = red[0]; __syncthreads();

  float acc = 0.f;
#pragma unroll
  for (int i = 0; i < 8; ++i) { float e = expf(x[i] - mx); x[i] = e; acc += e; }
  red[tid] = acc; __syncthreads();
  for (int s = 128; s > 0; s >>= 1) { if (tid < s) red[tid] += red[tid + s]; __syncthreads(); }
  float inv = 1.0f / red[0]; __syncthreads();

#pragma unroll
  for (int i = 0; i < 8; ++i) {
    int idx = tid + 256 * i;
    float wc = x[i] * inv;
    sh[idx] = g * wc + (1.0f - g) * wprev[idx];
  }
  __syncthreads();

  float wparr[8]; float acc2 = 0.f;
#pragma unroll
  for (int i = 0; i < 8; ++i) {
    int idx = tid + 256 * i;
    float wsv = s0 * sh[(idx + 1) & (N_ - 1)] + s1 * sh[idx] + s2 * sh[(idx - 1) & (N_ - 1)];
    float wpw = powf(wsv + EPS_, gamma);
    wparr[i] = wpw; acc2 += wpw;
  }
  red[tid] = acc2; __syncthreads();
  for (int s = 128; s > 0; s >>= 1) { if (tid < s) red[tid] += red[tid + s]; __syncthreads(); }
  float inv2 = 1.0f / red[0];
#pragma unroll
  for (int i = 0; i < 8; ++i) { int idx = tid + 256 * i; wout[idx] = wparr[i] * inv2; }
}

// ---------------------------------------------------------------------------
// Fused: reads einsum (4 heads) + apply write head 0 + cosine for write head 1.
// One read + one write of the memory tensor.
// ---------------------------------------------------------------------------
__global__ void reads_write0(const float* __restrict__ memIn, float* __restrict__ memOut,
                             const float* __restrict__ wRead, const float* __restrict__ wW0,
                             const float* __restrict__ erase, const float* __restrict__ addv,
                             const float* __restrict__ wp, const float* __restrict__ prm,
                             float* __restrict__ cosW1, float* __restrict__ readOut) {
  __shared__ float redc[8][4][64];
  int b = blockIdx.x;
  int lane = threadIdx.x & 31, wave = threadIdx.x >> 5;
  float2 ev = *(const float2*)(erase + (size_t)b * M_ + 2 * lane);
  float2 av = *(const float2*)(addv + (size_t)b * M_ + 2 * lane);
  float2 k1 = *(const float2*)(wp + (size_t)(1 * B_ + b) * PW_ + 2 * lane);
  float kn1 = prm[(size_t)(5 * B_ + b) * 8];
  const float2* mi = (const float2*)(memIn + (size_t)b * N_ * M_);
  float2* mo = (float2*)(memOut + (size_t)b * N_ * M_);
  float2 racc[4];
#pragma unroll
  for (int h = 0; h < 4; ++h) { racc[h].x = 0.f; racc[h].y = 0.f; }
  for (int n = wave; n < N_; n += 8) {
    __builtin_prefetch(mi + (size_t)(n + 64) * 32 + lane, 0, 0);
    float2 mv = mi[(size_t)n * 32 + lane];
    float w0n = wW0[(size_t)b * N_ + n];
#pragma unroll
    for (int h = 0; h < 4; ++h) {
      float wr = wRead[((size_t)h * B_ + b) * N_ + n];
      racc[h].x += wr * mv.x; racc[h].y += wr * mv.y;
    }
    float2 nv;
    nv.x = mv.x * (1.0f - w0n * ev.x) + w0n * av.x;
    nv.y = mv.y * (1.0f - w0n * ev.y) + w0n * av.y;
    mo[(size_t)n * 32 + lane] = nv;
    float d = nv.x * k1.x + nv.y * k1.y;
    float nr = nv.x * nv.x + nv.y * nv.y;
    d = wred(d); nr = wred(nr);
    if (lane == 0) cosW1[(size_t)b * N_ + n] = d / (sqrtf(nr) * kn1 + EPS_);
  }
#pragma unroll
  for (int h = 0; h < 4; ++h) {
    redc[wave][h][2 * lane] = racc[h].x;
    redc[wave][h][2 * lane + 1] = racc[h].y;
  }
  __syncthreads();
  int t = threadIdx.x, hh = t >> 6, m = t & 63;
  float s = 0.f;
#pragma unroll
  for (int w = 0; w < 8; ++w) s += redc[w][hh][m];
  readOut[(size_t)b * (R_ * M_) + hh * M_ + m] = s;
}

// ---------------------------------------------------------------------------
// Apply write head h in place; optionally compute cosine for head h+1.
// ---------------------------------------------------------------------------
__global__ void write_update(float* __restrict__ memIO, const float* __restrict__ wCur,
                             const float* __restrict__ erase, const float* __restrict__ addv,
                             int h, const float* __restrict__ wp,
                             const float* __restrict__ prm, float* __restrict__ cosNext) {
  int b = blockIdx.x;
  int lane = threadIdx.x & 31, wave = threadIdx.x >> 5;
  float2 ev = *(const float2*)(erase + ((size_t)h * B_ + b) * M_ + 2 * lane);
  float2 av = *(const float2*)(addv + ((size_t)h * B_ + b) * M_ + 2 * lane);
  float2 kv; kv.x = 0.f; kv.y = 0.f;
  float knn = 0.f;
  if (cosNext) {
    kv = *(const float2*)(wp + ((size_t)(h + 1) * B_ + b) * PW_ + 2 * lane);
    knn = prm[(size_t)((4 + h + 1) * B_ + b) * 8];
  }
  float2* mm = (float2*)(memIO + (size_t)b * N_ * M_);
  for (int n = wave; n < N_; n += 8) {
    __builtin_prefetch(mm + (size_t)(n + 64) * 32 + lane, 0, 0);
    float2 mv = mm[(size_t)n * 32 + lane];
    float wn = wCur[(size_t)b * N_ + n];
    float2 nv;
    nv.x = mv.x * (1.0f - wn * ev.x) + wn * av.x;
    nv.y = mv.y * (1.0f - wn * ev.y) + wn * av.y;
    mm[(size_t)n * 32 + lane] = nv;
    if (cosNext) {
      float d = nv.x * kv.x + nv.y * kv.y;
      float nr = nv.x * nv.x + nv.y * nv.y;
      d = wred(d); nr = wred(nr);
      if (lane == 0) cosNext[(size_t)b * N_ + n] = d / (sqrtf(nr) * knn + EPS_);
    }
  }
}

extern "C" void kernel_launch(void* const* d_in, const int* in_sizes, int n_in,
                              void* d_out, int out_size, void* d_ws, size_t ws_size,
                              hipStream_t stream) {
  const float* ext   = (const float*)d_in[0];
  const float* pread = (const float*)d_in[1];
  const float* prw   = (const float*)d_in[2];
  const float* pww   = (const float*)d_in[3];
  const float* mem   = (const float*)d_in[4];
  const float* ctrlW = (const float*)d_in[5];
  const float* ctrlb = (const float*)d_in[6];
  const float* readW = (const float*)d_in[7];
  const float* readb = (const float*)d_in[8];
  const float* writW = (const float*)d_in[9];
  const float* writb = (const float*)d_in[10];
  const float* outW  = (const float*)d_in[11];
  const float* outb  = (const float*)d_in[12];
  float* out = (float*)d_out;
  float* ws  = (float*)d_ws;

  // workspace layout (floats)
  float* ctrl  = ws;                         // 256*512
  float* rp    = ctrl + B_ * C_;             // 4*256*70
  float* wp    = rp + R_ * B_ * PR_;         // 4*256*198
  float* prm   = wp + WH_ * B_ * PW_;        // 8*256*8
  float* erase = prm + 8 * B_ * 8;           // 4*256*64
  float* addv  = erase + WH_ * B_ * M_;      // 4*256*64

  // output layout (floats)
  float* oNtm  = out;
  float* oMem  = oNtm + B_ * DO_;
  float* oRead = oMem + (size_t)B_ * N_ * M_;
  float* oNRW  = oRead + B_ * R_ * M_;
  float* oNWW  = oNRW + (size_t)R_ * B_ * N_;

  dim3 blk32(32);
  // ctrl = [external_input | past_readings] @ ctrl_W + ctrl_b
  wmma_gemm<<<dim3(C_ / 16, B_ / 16), blk32, 0, stream>>>(
      ext, DE_, DE_, pread, R_ * M_, ctrlW, C_, ctrlb, ctrl, C_, B_, C_, DE_ + R_ * M_);
  // read / write head projections
  for (int h = 0; h < R_; ++h)
    wmma_gemm<<<dim3((PR_ + 15) / 16, B_ / 16), blk32, 0, stream>>>(
        ctrl, C_, C_, ctrl, C_, readW + (size_t)h * C_ * PR_, PR_, readb + h * PR_,
        rp + (size_t)h * B_ * PR_, PR_, B_, PR_, C_);
  for (int h = 0; h < WH_; ++h)
    wmma_gemm<<<dim3((PW_ + 15) / 16, B_ / 16), blk32, 0, stream>>>(
        ctrl, C_, C_, ctrl, C_, writW + (size_t)h * C_ * PW_, PW_, writb + h * PW_,
        wp + (size_t)h * B_ * PW_, PW_, B_, PW_, C_);

  head_params<<<dim3(8), dim3(256), 0, stream>>>(rp, wp, prm, erase, addv);

  // cosines over original memory: 4 read heads + write head 0
  cos_pass<<<dim3(B_), dim3(256), 0, stream>>>(mem, rp, wp, prm, oNRW, oNWW);

  // addressing: read heads (in place in d_out new_read_w), then write head 0
  address_kernel<<<dim3(B_, R_), dim3(256), 0, stream>>>(oNRW, prw, prm, oNRW);
  address_kernel<<<dim3(B_, 1), dim3(256), 0, stream>>>(oNWW, pww, prm + 4 * B_ * 8, oNWW);

  // reads + apply write 0 + cosine for write 1
  reads_write0<<<dim3(B_), dim3(256), 0, stream>>>(
      mem, oMem, oNRW, oNWW, erase, addv, wp, prm, oNWW + (size_t)1 * B_ * N_, oRead);

  // sequential write heads 1..3
  for (int h = 1; h < WH_; ++h) {
    address_kernel<<<dim3(B_, 1), dim3(256), 0, stream>>>(
        oNWW + (size_t)h * B_ * N_, pww + (size_t)h * B_ * N_, prm + (4 + h) * B_ * 8,
        oNWW + (size_t)h * B_ * N_);
    float* cn = (h + 1 < WH_) ? (oNWW + (size_t)(h + 1) * B_ * N_) : nullptr;
    write_update<<<dim3(B_), dim3(256), 0, stream>>>(oMem, oNWW + (size_t)h * B_ * N_,
                                                     erase, addv, h, wp, prm, cn);
  }

  // ntm_out = [ctrl | read_out] @ out_W + out_b
  wmma_gemm<<<dim3(DO_ / 16, B_ / 16), blk32, 0, stream>>>(
      ctrl, C_, C_, oRead, R_ * M_, outW, DO_, outb, oNtm, DO_, B_, DO_, C_ + R_ * M_);
}